// ODE_func_gcn_46926812677053
// MI455X (gfx1250) — compile-verified
//
#include <hip/hip_runtime.h>
#include <hip/hip_bf16.h>
#include <math.h>

// ---------------- problem constants (from reference) ----------------
#define N_B    2
#define V_N    8000
#define NODES  (N_B * V_N)        // 16000
#define C_IN   16
#define C_HID  32
#define E_N    262144
#define K_W    125                // 5*5*5
#define S_TAPS 8                  // 2^3 spline taps
#define NPAIRS (E_N * S_TAPS)     // 2,097,152
#define TILES_PER_BLOCK 8         // 8 waves/block, one 16-row tile each
#define MAX_GROUPS (NPAIRS / (16 * TILES_PER_BLOCK) + K_W)

typedef float v2f __attribute__((ext_vector_type(2)));
typedef float v8f __attribute__((ext_vector_type(8)));
typedef unsigned int u32x4 __attribute__((ext_vector_type(4)));
typedef int i32x4 __attribute__((ext_vector_type(4)));
typedef int i32x8 __attribute__((ext_vector_type(8)));

__device__ __forceinline__ void atomic_add_f32(float* p, float v) {
  __hip_atomic_fetch_add(p, v, __ATOMIC_RELAXED, __HIP_MEMORY_SCOPE_AGENT);
}

// ---------------- 1. spline basis + weight indices ----------------
__global__ void basis_kernel(const float* __restrict__ attr,
                             float* __restrict__ basis,
                             int* __restrict__ widx) {
  int e = blockIdx.x * blockDim.x + threadIdx.x;
  if (e >= E_N) return;
  float f[3]; int li[3];
#pragma unroll
  for (int d = 0; d < 3; ++d) {
    float v = attr[e * 3 + d] * 4.0f;   // pseudo * (kernel-1)
    float lo = floorf(v);
    f[d] = v - lo;
    li[d] = (int)lo;
  }
  const int stride[3] = {1, 5, 25};
#pragma unroll
  for (int s = 0; s < S_TAPS; ++s) {
    float b = 1.0f; int flat = 0;
#pragma unroll
    for (int d = 0; d < 3; ++d) {
      int bit = (s >> d) & 1;
      b *= bit ? f[d] : (1.0f - f[d]);
      int idx = li[d] + bit;
      idx = idx < 0 ? 0 : (idx > 4 ? 4 : idx);
      flat += idx * stride[d];
    }
    basis[e * S_TAPS + s] = b;
    widx[e * S_TAPS + s]  = flat;
  }
}

// ---------------- 2. bucket histogram ----------------
__global__ void count_kernel(const int* __restrict__ widx, int* __restrict__ counts) {
  int p = blockIdx.x * blockDim.x + threadIdx.x;
  if (p >= NPAIRS) return;
  atomicAdd(&counts[widx[p]], 1);
}

// ---------------- 3. scan ----------------
// meta (ints): [0..127]=counts [128..255]=offsets [256..383]=cursor
//              [384..511]=tile_off [512]=total_tiles
//              [640..767]=group_off [768]=total_groups
__global__ void scan_kernel(int* __restrict__ meta) {
  if (threadIdx.x != 0 || blockIdx.x != 0) return;
  int* counts    = meta;
  int* offsets   = meta + 128;
  int* cursor    = meta + 256;
  int* tile_off  = meta + 384;
  int* group_off = meta + 640;
  int run = 0, trun = 0, grun = 0;
  for (int k = 0; k < K_W; ++k) {
    offsets[k]   = run;
    cursor[k]    = run;
    tile_off[k]  = trun;
    group_off[k] = grun;
    int c = counts[k];
    int tiles = (c + 15) >> 4;
    run  += c;
    trun += tiles;
    grun += (tiles + TILES_PER_BLOCK - 1) / TILES_PER_BLOCK;
  }
  offsets[K_W]   = run;
  tile_off[K_W]  = trun;
  group_off[K_W] = grun;
  meta[512]      = trun;
  meta[768]      = grun;
}

// ---------------- 4. scatter pairs into bucket order ----------------
__global__ void scatter_kernel(const int* __restrict__ widx,
                               int* __restrict__ cursor,
                               int* __restrict__ pairs) {
  int p = blockIdx.x * blockDim.x + threadIdx.x;
  if (p >= NPAIRS) return;
  int k = widx[p];
  int pos = atomicAdd(&cursor[k], 1);
  pairs[pos] = p;                        // p = e*8 + s
}

// ---------------- 5. degree ----------------
__global__ void deg_kernel(const int* __restrict__ edge_index, float* __restrict__ deg) {
  int e = blockIdx.x * blockDim.x + threadIdx.x;
  if (e >= E_N) return;
  atomic_add_f32(&deg[edge_index[E_N + e]], 1.0f);
}

// ---------------- 6. grouped GEMM: fp32 WMMA + TDM weight staging ----------------
// Block = 8 waves = 8 consecutive 16-row tiles of ONE weight bucket. W[bucket]
// (CIN x COUT fp32, contiguous) is staged into LDS once per block by the Tensor
// Data Mover (TENSORcnt), then each wave independently: gathers 16 basis-scaled
// source rows into its private LDS slice, runs the V_WMMA_F32_16X16X4_F32
// K-loop (two N-halves for COUT=32), and scatters rows into agg[dst] with
// global_atomic_add_f32.
template <int CIN, int COUT>
__global__ __launch_bounds__(256) void spline_gemm_kernel(
    const int* __restrict__ pairs, const float* __restrict__ basis,
    const int* __restrict__ edge_index, const float* __restrict__ xin,
    const float* __restrict__ W, float* __restrict__ agg,
    const int* __restrict__ meta) {
  const int* offsets   = meta + 128;
  const int* group_off = meta + 640;
  int gid = blockIdx.x;
  if (gid >= meta[768]) return;          // block-uniform exit (before any barrier)

  int bucket = 0;
#pragma unroll 1
  for (int k = 0; k < K_W; ++k) {
    if (gid < group_off[k + 1]) { bucket = k; break; }
  }
  int group = gid - group_off[bucket];
  int pB    = offsets[bucket];
  int pend  = offsets[bucket + 1];

  __shared__ float sW[CIN][COUT];
  __shared__ float sY[TILES_PER_BLOCK][16][CIN];
  __shared__ int   sDst[TILES_PER_BLOCK][16];

  int t    = threadIdx.x;
  int wave = t >> 5;
  int lane = t & 31;

  // ---- stage W[bucket] -> LDS, once per block ----
#if __has_builtin(__builtin_amdgcn_tensor_load_to_lds)
  if (wave == 0) {
    // Tensor DMA descriptor (CDNA5 ISA): 1-D tile of CIN*COUT fp32.
    const unsigned NEL = (unsigned)(CIN * COUT);
    unsigned lds = (unsigned)(size_t)&sW[0][0];
    unsigned long long ga =
        (unsigned long long)(size_t)(W + (size_t)bucket * CIN * COUT);
    u32x4 g0;
    g0[0] = 1u;                                   // count=1, user descriptor
    g0[1] = lds;                                  // lds_addr
    g0[2] = (unsigned)ga;                         // global_addr[31:0]
    g0[3] = (unsigned)((ga >> 32) & 0x1FFFFFFu)   // global_addr[56:32]
            | (2u << 30);                         // type = 2 ("image")
    i32x8 g1;
    g1[0] = (int)(2u << 16);                      // data_size = 2 (4 bytes)
    g1[1] = (int)((NEL & 0xFFFFu) << 16);         // tensor_dim0[15:0]
    g1[2] = (int)((NEL >> 16) | (1u << 16));      // tensor_dim0[31:16], tensor_dim1=1
    g1[3] = (int)((NEL & 0xFFFFu) << 16);         // tile_dim0 = NEL
    g1[4] = 1;                                    // tile_dim1 = 1
    g1[5] = (int)NEL;                             // tensor_dim0_stride
    g1[6] = 0;
    g1[7] = 0;
    i32x4 gz4 = {0, 0, 0, 0};
    i32x8 gz8 = {0, 0, 0, 0, 0, 0, 0, 0};
    // 6-arg toolchain form: (g0, g1, grp2, grp3, extra, cpol)
    __builtin_amdgcn_tensor_load_to_lds(g0, g1, gz4, gz4, gz8, 0);
    __builtin_amdgcn_s_wait_tensorcnt(0);
  }
#else
  for (int i = t; i < CIN * COUT; i += 256)
    sW[i / COUT][i % COUT] = W[bucket * CIN * COUT + i];
#endif
  __syncthreads();

  // ---- per-wave tile ----
  int tile   = group * TILES_PER_BLOCK + wave;
  int pstart = pB + tile * 16;
  if (pstart < pend) {                   // wave-uniform; EXEC all-1s inside
    float (*Y)[CIN] = sY[wave];
    int*   D        = sDst[wave];

#pragma unroll 1
    for (int r = 0; r < 16; ++r) {
      int p = pstart + r;
      if (p < pend) {
        int packed  = pairs[p];
        int e       = packed >> 3;
        float scale = basis[packed];
        int srcn    = edge_index[e];
        if (lane < CIN) Y[r][lane] = scale * xin[srcn * CIN + lane];
        if (lane == 0)  D[r] = edge_index[E_N + e];
      } else {
        if (lane < CIN) Y[r][lane] = 0.0f;
        if (lane == 0)  D[r] = -1;
      }
    }
    // same-wave LDS RAW: DS pipeline is in-order; compiler inserts dscnt waits.

    int half = lane >> 4;      // 0: K {0,1} / rows M0-7 ; 1: K {2,3} / rows M8-15
    int l16  = lane & 15;

    v8f acc0 = {};             // N = 0..15
    v8f acc1 = {};             // N = 16..31 (COUT==32 only)
#pragma unroll
    for (int kk = 0; kk < CIN; kk += 4) {
      int kA = kk + 2 * half;
      v2f a;  a.x  = Y[l16][kA];         a.y  = Y[l16][kA + 1];
      v2f b0; b0.x = sW[kA][l16];        b0.y = sW[kA + 1][l16];
      acc0 = __builtin_amdgcn_wmma_f32_16x16x4_f32(false, a, false, b0,
                                                   (short)0, acc0, false, false);
      if (COUT == 32) {
        v2f b1; b1.x = sW[kA][l16 + 16]; b1.y = sW[kA + 1][l16 + 16];
        acc1 = __builtin_amdgcn_wmma_f32_16x16x4_f32(false, a, false, b1,
                                                     (short)0, acc1, false, false);
      }
    }

#pragma unroll
    for (int r = 0; r < 8; ++r) {
      int row = r + 8 * half;  // C/D layout: VGPR r, lanes>=16 hold M=r+8
      int d = D[row];
      if (d >= 0) {
        atomic_add_f32(&agg[d * COUT + l16], acc0[r]);
        if (COUT == 32) atomic_add_f32(&agg[d * COUT + 16 + l16], acc1[r]);
      }
    }
  }
}

// ---------------- 7. finalize: deg-normalize + root GEMM + bias + activation ----
template <int CIN, int COUT, int ACT>   // ACT 0 = ELU, 1 = tanh
__global__ void finalize_kernel(const float* __restrict__ agg,
                                const float* __restrict__ deg,
                                const float* __restrict__ xin,
                                const float* __restrict__ root,
                                const float* __restrict__ bias,
                                float* __restrict__ out) {
  int idx = blockIdx.x * blockDim.x + threadIdx.x;
  if (idx >= NODES * COUT) return;
  int n = idx / COUT, o = idx % COUT;
  float acc = agg[idx] / fmaxf(deg[n], 1.0f);
  const float* xr = xin + n * CIN;
#pragma unroll
  for (int i = 0; i < CIN; ++i) acc += xr[i] * root[i * COUT + o];
  acc += bias[o];
  if (ACT == 0) acc = acc > 0.0f ? acc : (expf(acc) - 1.0f);
  else          acc = tanhf(acc);
  out[idx] = acc;
}

// ---------------- host side ----------------
extern "C" void kernel_launch(void* const* d_in, const int* in_sizes, int n_in,
                              void* d_out, int out_size, void* d_ws, size_t ws_size,
                              hipStream_t stream) {
  const float* x         = (const float*)d_in[1];
  const int*   edge_index= (const int*)  d_in[2];
  const float* edge_attr = (const float*)d_in[3];
  const float* W_in      = (const float*)d_in[4];
  const float* root_in   = (const float*)d_in[5];
  const float* b_in      = (const float*)d_in[6];
  const float* W_h       = (const float*)d_in[7];
  const float* root_h    = (const float*)d_in[8];
  const float* b_h       = (const float*)d_in[9];
  const float* W_out     = (const float*)d_in[10];
  const float* root_out  = (const float*)d_in[11];
  const float* b_out     = (const float*)d_in[12];
  float* out = (float*)d_out;

  // workspace layout
  char* ws = (char*)d_ws;
  float* basis = (float*)(ws);                              // 8 MB
  int*   widx  = (int*)  (ws + 8388608);                    // 8 MB
  int*   pairs = (int*)  (ws + 16777216);                   // 8 MB
  int*   meta  = (int*)  (ws + 25165824);                   // 4 KB
  float* deg   = (float*)(ws + 25169920);                   // 64 KB slot
  float* agg   = (float*)(ws + 25235456);                   // 2 MB
  float* h0    = (float*)(ws + 27332608);                   // 2 MB
  float* h1    = (float*)(ws + 29429760);                   // 2 MB

  (void)hipMemsetAsync(meta, 0, 4096, stream);
  (void)hipMemsetAsync(deg, 0, NODES * sizeof(float), stream);

  basis_kernel  <<<(E_N    + 255) / 256, 256, 0, stream>>>(edge_attr, basis, widx);
  count_kernel  <<<(NPAIRS + 255) / 256, 256, 0, stream>>>(widx, meta);
  deg_kernel    <<<(E_N    + 255) / 256, 256, 0, stream>>>(edge_index, deg);
  scan_kernel   <<<1, 32, 0, stream>>>(meta);
  scatter_kernel<<<(NPAIRS + 255) / 256, 256, 0, stream>>>(widx, meta + 256, pairs);

  // layer 0: conv 16 -> 32, ELU
  (void)hipMemsetAsync(agg, 0, NODES * 32 * sizeof(float), stream);
  spline_gemm_kernel<C_IN, C_HID><<<MAX_GROUPS, 256, 0, stream>>>(
      pairs, basis, edge_index, x, W_in, agg, meta);
  finalize_kernel<C_IN, C_HID, 0><<<(NODES * C_HID + 255) / 256, 256, 0, stream>>>(
      agg, deg, x, root_in, b_in, h0);

  // hidden layer 0: conv 32 -> 32, ELU
  (void)hipMemsetAsync(agg, 0, NODES * 32 * sizeof(float), stream);
  spline_gemm_kernel<C_HID, C_HID><<<MAX_GROUPS, 256, 0, stream>>>(
      pairs, basis, edge_index, h0, W_h, agg, meta);
  finalize_kernel<C_HID, C_HID, 0><<<(NODES * C_HID + 255) / 256, 256, 0, stream>>>(
      agg, deg, h0, root_h, b_h, h1);

  // hidden layer 1: conv 32 -> 32, ELU
  (void)hipMemsetAsync(agg, 0, NODES * 32 * sizeof(float), stream);
  spline_gemm_kernel<C_HID, C_HID><<<MAX_GROUPS, 256, 0, stream>>>(
      pairs, basis, edge_index, h1, W_h + K_W * C_HID * C_HID, agg, meta);
  finalize_kernel<C_HID, C_HID, 0><<<(NODES * C_HID + 255) / 256, 256, 0, stream>>>(
      agg, deg, h1, root_h + C_HID * C_HID, b_h + C_HID, h0);

  // output layer: conv 32 -> 16, tanh
  (void)hipMemsetAsync(agg, 0, NODES * 16 * sizeof(float), stream);
  spline_gemm_kernel<C_HID, C_IN><<<MAX_GROUPS, 256, 0, stream>>>(
      pairs, basis, edge_index, h0, W_out, agg, meta);
  finalize_kernel<C_HID, C_IN, 1><<<(NODES * C_IN + 255) / 256, 256, 0, stream>>>(
      agg, deg, h0, root_out, b_out, out);
}